// EquivariantAttention_63419487093209
// MI455X (gfx1250) — compile-verified
//
#include <hip/hip_runtime.h>
#include <hip/hip_bf16.h>
#include <float.h>

#define BB 2
#define NN 4096
#define DIMF 256
#define HEADS 4
#define DH 64
#define KNN 32
#define MHID 16
// ln(10000)/16
#define LOG1E4_OVER_16 0.5756462732485115f

typedef float v2f __attribute__((ext_vector_type(2)));
typedef float v8f __attribute__((ext_vector_type(8)));

__device__ __forceinline__ bool lex_less(float ad, int ai, float bd, int bi) {
    return (ad < bd) || (ad == bd && ai < bi);
}

// ---------------------------------------------------------------------------
// Kernel 1: exact kNN (top-32 by distance, stable tie-break by index).
// One wave (32 lanes) per query point. Each lane scans N/32 = 128 candidates
// keeping a sorted in-register top-32, then a 32-way LDS merge selects the
// global top-32 in ascending (dist, idx) order == jax.lax.top_k(-dist, 32).
// ---------------------------------------------------------------------------
__global__ void knn_kernel(const float* __restrict__ coors,
                           int* __restrict__ nb_idx,
                           float* __restrict__ nb_dist) {
    const int row = blockIdx.x;          // b*N + i
    const int b = row / NN;
    const int lane = threadIdx.x;        // 0..31

    const float cx = coors[(size_t)row * 3 + 0];
    const float cy = coors[(size_t)row * 3 + 1];
    const float cz = coors[(size_t)row * 3 + 2];
    const float* cb = coors + (size_t)b * NN * 3;

    float bd[KNN];
    int   bi[KNN];
#pragma unroll
    for (int r = 0; r < KNN; ++r) { bd[r] = FLT_MAX; bi[r] = 0x7fffffff; }

    for (int j = lane; j < NN; j += 32) {
        const float dx = cx - cb[(size_t)j * 3 + 0];
        const float dy = cy - cb[(size_t)j * 3 + 1];
        const float dz = cz - cb[(size_t)j * 3 + 2];
        const float d2 = dx * dx + dy * dy + dz * dz;
        if (lex_less(d2, j, bd[KNN - 1], bi[KNN - 1])) {
            // branchless sorted insert (all indices compile-time)
#pragma unroll
            for (int t = KNN - 1; t >= 0; --t) {
                const bool lt_t = lex_less(d2, j, bd[t], bi[t]);
                if (lt_t) {
                    const bool lt_prev = (t == 0) ? false : lex_less(d2, j, bd[t - 1], bi[t - 1]);
                    if (lt_prev) { bd[t] = bd[t - 1]; bi[t] = bi[t - 1]; }
                    else         { bd[t] = d2;        bi[t] = j;         }
                }
            }
        }
    }

    __shared__ float sd[32 * KNN];
    __shared__ int   si[32 * KNN];
#pragma unroll
    for (int r = 0; r < KNN; ++r) { sd[lane * KNN + r] = bd[r]; si[lane * KNN + r] = bi[r]; }
    __syncthreads();

    // 32-way merge: each step selects the global minimum of the 32 list heads.
    int p = 0;
    float out_d = FLT_MAX;
    int   out_i = 0;
    for (int s = 0; s < KNN; ++s) {
        float md = (p < KNN) ? sd[lane * KNN + p] : FLT_MAX;
        int   mi = (p < KNN) ? si[lane * KNN + p] : 0x7fffffff;
        int   ml = lane;
        for (int off = 16; off; off >>= 1) {
            const float od = __shfl_xor(md, off);
            const int   oi = __shfl_xor(mi, off);
            const int   ol = __shfl_xor(ml, off);
            if (lex_less(od, oi, md, mi)) { md = od; mi = oi; ml = ol; }
        }
        if (ml == lane) ++p;               // winner consumes its head
        if (lane == s) { out_d = md; out_i = mi; }
    }
    nb_idx [(size_t)row * KNN + lane] = out_i;
    nb_dist[(size_t)row * KNN + lane] = sqrtf(out_d);
}

// ---------------------------------------------------------------------------
// Kernel 2/4: fp32 GEMM on the CDNA5 matrix pipe: V_WMMA_F32_16X16X4_F32.
// One wave per 16x16 output tile, 4 tiles (waves) per block along N.
// A 16x4 f32 layout: lanes 0-15 -> K=0,1 ; lanes 16-31 -> K=2,3 (ISA 7.12.2).
// D layout: VGPR r -> row m0 + r + 8*half, col n0 + (lane&15).
// ---------------------------------------------------------------------------
__global__ void wmma_gemm_f32(const float* __restrict__ A,
                              const float* __restrict__ B,
                              const float* __restrict__ bias,
                              float* __restrict__ C,
                              int M, int N, int K) {
    const int lane = threadIdx.x;               // 0..31
    const int n0 = (blockIdx.x * 4 + threadIdx.y) * 16;
    const int m0 = blockIdx.y * 16;
    if (n0 >= N || m0 >= M) return;             // wave-uniform -> EXEC stays all-1s

    const int half = lane >> 4;
    const int ln   = lane & 15;
    const float* arow = A + (size_t)(m0 + ln) * K;

    v8f acc = {0.f, 0.f, 0.f, 0.f, 0.f, 0.f, 0.f, 0.f};
    for (int k = 0; k < K; k += 4) {
        const int ka = k + 2 * half;
        v2f a, bb;
        a.x  = arow[ka];
        a.y  = arow[ka + 1];
        bb.x = B[(size_t)ka * N + n0 + ln];
        bb.y = B[(size_t)(ka + 1) * N + n0 + ln];
        acc = __builtin_amdgcn_wmma_f32_16x16x4_f32(
            false, a, false, bb, (short)0, acc, false, false);
    }

    const float bv = bias ? bias[n0 + ln] : 0.f;
#pragma unroll
    for (int r = 0; r < 8; ++r) {
        C[(size_t)(m0 + r + 8 * half) * N + n0 + ln] = acc[r] + bv;
    }
}

// ---------------------------------------------------------------------------
// Kernel 3: gathered attention + rotary + softmax + coordinate branch.
// Block = (32, 4): wave h handles head h for query (b, i) = blockIdx.
// Phase 1: lane j = neighbor j -> rotary(k), qk dot, softmax (q rotary is
//          identity since t=0). Phase 2: lane l = dim pair (2l, 2l+1) ->
//          rotary(v), attn-weighted sum. Phase 3 (wave 0): coord MLP + output.
// Note: reference layernorm is over a singleton axis -> phase == ln_b exactly.
// ---------------------------------------------------------------------------
__global__ void attn_kernel(const float* __restrict__ qkv,
                            const int* __restrict__ nb_idx,
                            const float* __restrict__ nb_dist,
                            const float* __restrict__ coors,
                            const float* __restrict__ w_c1,
                            const float* __restrict__ b_c1,
                            const float* __restrict__ w_c2,
                            const float* __restrict__ b_c2,
                            const float* __restrict__ ln_b,
                            float* __restrict__ attn_out,
                            float* __restrict__ coors_out) {
    const int row = blockIdx.x;            // b*N + i
    const int b = row / NN;
    const int lane = threadIdx.x;          // 0..31
    const int h = threadIdx.y;             // 0..3

    __shared__ float s_qk[KNN * HEADS];    // [j*4 + h], pre-softmax scaled logits
    __shared__ float s_attn[HEADS * KNN];  // [h*32 + j]
    __shared__ float s_t[KNN];
    __shared__ int   s_idx[KNN];

    // ---- phase 1: logits + softmax (lane = neighbor) ----
    const int   j    = lane;
    const int   nidx = nb_idx [(size_t)row * KNN + j];
    const float t    = nb_dist[(size_t)row * KNN + j] * 100.f;
    if (h == 0) { s_idx[j] = nidx; s_t[j] = t; }

    const float* qp = qkv + (size_t)row * 768 + h * DH;
    const float* kp = qkv + (size_t)((size_t)b * NN + nidx) * 768 + 256 + h * DH;

    float qk = 0.f;
#pragma unroll
    for (int p = 0; p < 16; ++p) {         // rotary half (dims 0..31)
        const float f = t * __expf(-(float)p * LOG1E4_OVER_16);
        float sn, cs;
        __sincosf(f, &sn, &cs);
        const float k0 = kp[2 * p], k1 = kp[2 * p + 1];
        qk += qp[2 * p]     * (k0 * cs - k1 * sn)
            + qp[2 * p + 1] * (k1 * cs + k0 * sn);
    }
#pragma unroll
    for (int d = 32; d < 64; ++d) qk += qp[d] * kp[d];   // pass-through half
    qk *= 0.125f;                                        // dh^-0.5
    s_qk[j * HEADS + h] = qk;

    float m = qk;
    for (int off = 16; off; off >>= 1) m = fmaxf(m, __shfl_xor(m, off));
    const float e = __expf(qk - m);
    float ssum = e;
    for (int off = 16; off; off >>= 1) ssum += __shfl_xor(ssum, off);
    s_attn[h * KNN + j] = e / ssum;
    __syncthreads();

    // ---- phase 2: attn @ rotary(v) (lane = dim pair) ----
    const int d0 = 2 * lane, d1 = 2 * lane + 1;
    const float invf = __expf(-(float)lane * LOG1E4_OVER_16);  // valid for lane<16
    float acc0 = 0.f, acc1 = 0.f;
    for (int jj = 0; jj < KNN; ++jj) {
        const float a = s_attn[h * KNN + jj];
        const float* vp = qkv + (size_t)((size_t)b * NN + s_idx[jj]) * 768 + 512 + h * DH;
        const float v0 = vp[d0], v1 = vp[d1];
        if (lane < 16) {
            float sn, cs;
            __sincosf(s_t[jj] * invf, &sn, &cs);
            acc0 += a * (v0 * cs - v1 * sn);
            acc1 += a * (v1 * cs + v0 * sn);
        } else {
            acc0 += a * v0;
            acc1 += a * v1;
        }
    }
    attn_out[(size_t)row * DIMF + h * DH + d0] = acc0;
    attn_out[(size_t)row * DIMF + h * DH + d1] = acc1;

    // ---- phase 3: coordinate branch (wave 0, lane = neighbor) ----
    if (h == 0) {
        float q0 = s_qk[j * HEADS + 0], q1 = s_qk[j * HEADS + 1];
        float q2 = s_qk[j * HEADS + 2], q3 = s_qk[j * HEADS + 3];
        float cw = b_c2[0];
#pragma unroll
        for (int mh = 0; mh < MHID; ++mh) {
            float x = b_c1[mh] + q0 * w_c1[0 * MHID + mh] + q1 * w_c1[1 * MHID + mh]
                               + q2 * w_c1[2 * MHID + mh] + q3 * w_c1[3 * MHID + mh];
            const float g = 0.5f * x * (1.f + erff(x * 0.70710678118654752f));  // exact gelu
            cw += g * w_c2[mh];
        }
        const int nb = s_idx[j];
        float rx = coors[(size_t)row * 3 + 0] - coors[(size_t)((size_t)b * NN + nb) * 3 + 0];
        float ry = coors[(size_t)row * 3 + 1] - coors[(size_t)((size_t)b * NN + nb) * 3 + 1];
        float rz = coors[(size_t)row * 3 + 2] - coors[(size_t)((size_t)b * NN + nb) * 3 + 2];
        const float nrm = sqrtf(rx * rx + ry * ry + rz * rz);
        const float inv = 1.f / fmaxf(nrm, 1e-8f);
        const float w = cw * ln_b[0] * inv;  // phase == ln_b (singleton-axis layernorm)
        float vx = w * rx, vy = w * ry, vz = w * rz;
        for (int off = 16; off; off >>= 1) {
            vx += __shfl_xor(vx, off);
            vy += __shfl_xor(vy, off);
            vz += __shfl_xor(vz, off);
        }
        if (lane == 0) {
            coors_out[(size_t)row * 3 + 0] = vx;
            coors_out[(size_t)row * 3 + 1] = vy;
            coors_out[(size_t)row * 3 + 2] = vz;
        }
    }
}

extern "C" void kernel_launch(void* const* d_in, const int* in_sizes, int n_in,
                              void* d_out, int out_size, void* d_ws, size_t ws_size,
                              hipStream_t stream) {
    const float* feats = (const float*)d_in[0];
    const float* coors = (const float*)d_in[1];
    const float* w_qkv = (const float*)d_in[2];
    const float* w_out = (const float*)d_in[3];
    const float* b_out = (const float*)d_in[4];
    const float* w_c1  = (const float*)d_in[5];
    const float* b_c1  = (const float*)d_in[6];
    const float* w_c2  = (const float*)d_in[7];
    const float* b_c2  = (const float*)d_in[8];
    // d_in[9] = ln_g (multiplies an exact zero in the reference -> unused)
    const float* ln_b  = (const float*)d_in[10];

    const int rows = BB * NN;                       // 8192

    // workspace carve-up
    char* ws = (char*)d_ws;
    int*   nb_idx   = (int*)ws;                     ws += (size_t)rows * KNN * 4;
    float* nb_dist  = (float*)ws;                   ws += (size_t)rows * KNN * 4;
    float* qkv      = (float*)ws;                   ws += (size_t)rows * 768 * 4;
    float* attn_out = (float*)ws;                   ws += (size_t)rows * DIMF * 4;

    float* out       = (float*)d_out;               // (2,4096,256)
    float* coors_out = out + (size_t)rows * DIMF;   // (2,4096,3)

    // 1) exact kNN
    knn_kernel<<<dim3(rows), dim3(32), 0, stream>>>(coors, nb_idx, nb_dist);

    // 2) qkv = feats @ w_qkv   (8192 x 768, K=256) — f32 WMMA
    wmma_gemm_f32<<<dim3(768 / 64, rows / 16), dim3(32, 4), 0, stream>>>(
        feats, w_qkv, nullptr, qkv, rows, 768, DIMF);

    // 3) rotary + attention + coordinate branch
    attn_kernel<<<dim3(rows), dim3(32, 4), 0, stream>>>(
        qkv, nb_idx, nb_dist, coors, w_c1, b_c1, w_c2, b_c2, ln_b,
        attn_out, coors_out);

    // 4) out = attn_out @ w_out + b_out  (8192 x 256, K=256) — f32 WMMA
    wmma_gemm_f32<<<dim3(DIMF / 64, rows / 16), dim3(32, 4), 0, stream>>>(
        attn_out, w_out, b_out, out, rows, DIMF, DIMF);
}